// GraphEncoder_90572270338522
// MI455X (gfx1250) — compile-verified
//
#include <hip/hip_runtime.h>
#include <math.h>
#include <stddef.h>

// Problem constants (from reference)
#define BB  8
#define NN  512
#define KKE 16      // edges per node
#define DD  128
#define DE  64
#define FF  256
#define HH  8
#define HDIM 16
#define LLAY 12
#define LN_EPS 1e-5f

typedef float v2f __attribute__((ext_vector_type(2)));
typedef float v8f __attribute__((ext_vector_type(8)));

__device__ __forceinline__ float gelu_f(float x) {
    return 0.5f * x * (1.0f + erff(x * 0.70710678118654752f));
}

__device__ __forceinline__ v8f wmma4(v2f a, v2f b, v8f c) {
    // D = A(16x4 f32) x B(4x16 f32) + C(16x16 f32)
    return __builtin_amdgcn_wmma_f32_16x16x4_f32(false, a, false, b, (short)0, c, false, false);
}

// ---------------------------------------------------------------------------
// Generic WMMA GEMM: Out[M x Ncols] = A[M x Kd] @ Bm[Kd x Ncols] (+bias)(+resid)
// One wave per 16x16 output tile. scale applied as (acc+bias)*scale. act=1 -> GELU.
// ---------------------------------------------------------------------------
__global__ __launch_bounds__(32)
void wmma_gemm_kernel(const float* __restrict__ A, const float* __restrict__ Bm,
                      const float* __restrict__ bias, const float* __restrict__ resid,
                      float* __restrict__ Out, int Kd, int Ncols, float scale, int act)
{
    const int lane = threadIdx.x;
    const int g  = lane >> 4;     // half-wave group
    const int lm = lane & 15;
    const long arow = (long)blockIdx.x * 16 + lm;
    const int  bcol = blockIdx.y * 16 + lm;
    const float* Ap = A + arow * Kd;

    v8f acc = {};
    for (int k0 = 0; k0 < Kd; k0 += 4) {
        v2f a = { Ap[k0 + 2*g], Ap[k0 + 2*g + 1] };
        v2f b = { Bm[(long)(k0 + 2*g    ) * Ncols + bcol],
                  Bm[(long)(k0 + 2*g + 1) * Ncols + bcol] };
        acc = wmma4(a, b, acc);
    }
#pragma unroll
    for (int vv = 0; vv < 8; ++vv) {
        const long row = (long)blockIdx.x * 16 + vv + 8*g;
        float val = (acc[vv] + bias[bcol]) * scale;
        if (resid) val += resid[row * Ncols + bcol];
        if (act)   val = gelu_f(val);
        Out[row * Ncols + bcol] = val;
    }
}

// ---------------------------------------------------------------------------
// Flash-style attention: one wave per (b, h, 16-row tile). Single pass over
// 32 key tiles with online softmax. Saves per-row max & sum for later exact
// recomputation of mean-head attention at gathered (j,n) pairs.
// q is pre-scaled by HD^-0.5. bias is (B,N,N).
// ---------------------------------------------------------------------------
__global__ __launch_bounds__(32)
void attn_flash_kernel(const float* __restrict__ q, const float* __restrict__ k,
                       const float* __restrict__ v, const float* __restrict__ bias,
                       float* __restrict__ o, float* __restrict__ Mrow,
                       float* __restrict__ Lrow)
{
    const int b = blockIdx.z, h = blockIdx.y, rt = blockIdx.x;
    const int lane = threadIdx.x;
    const int g  = lane >> 4;
    const int lm = lane & 15;

    __shared__ float plds[16][17];

    // Q tile in A-layout registers: row = lm, K (head dim) = 4*s + 2*g + j
    float qa[8];
    {
        const int qrow = rt * 16 + lm;
        const float* qp = q + ((long)(b * NN + qrow)) * DD + h * HDIM;
#pragma unroll
        for (int s = 0; s < 4; ++s) {
            qa[2*s]   = qp[4*s + 2*g];
            qa[2*s+1] = qp[4*s + 2*g + 1];
        }
    }

    float mr[8], lr[8];
    v8f oacc = {};
#pragma unroll
    for (int vv = 0; vv < 8; ++vv) { mr[vv] = -INFINITY; lr[vv] = 0.0f; }

    for (int ct = 0; ct < NN / 16; ++ct) {
        // S tile = Q (16xHD) @ K^T (HDx16)
        v8f sacc = {};
        {
            const float* kp = k + ((long)(b * NN + ct * 16 + lm)) * DD + h * HDIM;
#pragma unroll
            for (int s = 0; s < 4; ++s) {
                v2f a = { qa[2*s], qa[2*s+1] };
                v2f bb = { kp[4*s + 2*g], kp[4*s + 2*g + 1] };
                sacc = wmma4(a, bb, sacc);
            }
        }
        // bias + online softmax update (rows spread across 16-lane halves)
#pragma unroll
        for (int vv = 0; vv < 8; ++vv) {
            const int r = rt * 16 + vv + 8*g;
            const int c = ct * 16 + lm;
            float sv = sacc[vv] + bias[((long)(b * NN + r)) * NN + c];
            float mx = sv;
#pragma unroll
            for (int off = 1; off < 16; off <<= 1) mx = fmaxf(mx, __shfl_xor(mx, off));
            const float nm = fmaxf(mr[vv], mx);
            const float p  = expf(sv - nm);
            float rs = p;
#pragma unroll
            for (int off = 1; off < 16; off <<= 1) rs += __shfl_xor(rs, off);
            const float corr = expf(mr[vv] - nm);
            lr[vv]   = lr[vv] * corr + rs;
            oacc[vv] = oacc[vv] * corr;
            mr[vv]   = nm;
            plds[vv + 8*g][lm] = p;
        }
        __syncthreads();
        // O += P (16x16) @ Vtile (16xHD), P re-read from LDS in A-layout
        {
            const float* vp = v + ((long)(b * NN + ct * 16)) * DD + h * HDIM + lm;
#pragma unroll
            for (int s = 0; s < 4; ++s) {
                v2f a  = { plds[lm][4*s + 2*g], plds[lm][4*s + 2*g + 1] };
                v2f bb = { vp[(4*s + 2*g) * DD], vp[(4*s + 2*g + 1) * DD] };
                oacc = wmma4(a, bb, oacc);
            }
        }
        __syncthreads();
    }

#pragma unroll
    for (int vv = 0; vv < 8; ++vv) {
        const int r = rt * 16 + vv + 8*g;
        o[((long)(b * NN + r)) * DD + h * HDIM + lm] = oacc[vv] / lr[vv];
    }
    if (lm == 0) {
#pragma unroll
        for (int vv = 0; vv < 8; ++vv) {
            const int r = rt * 16 + vv + 8*g;
            Mrow[(b * HH + h) * NN + r] = mr[vv];
            Lrow[(b * HH + h) * NN + r] = lr[vv];
        }
    }
}

// ---------------------------------------------------------------------------
// attn_bias scatter: bias[b, n, edge_index[b,n,k]] = mean_d e[b,n,k,:]
// ---------------------------------------------------------------------------
__global__ void bias_scatter_kernel(const float* __restrict__ e, const int* __restrict__ eidx,
                                    float* __restrict__ bias)
{
    const int t = blockIdx.x * 256 + threadIdx.x;
    if (t >= BB * NN * KKE) return;
    const int b = t / (NN * KKE);
    const int n = (t - b * NN * KKE) / KKE;
    const float* ep = e + (long)t * DE;
    float s = 0.0f;
    for (int d = 0; d < DE; ++d) s += ep[d];
    bias[((long)(b * NN + n)) * NN + eidx[t]] = s * (1.0f / DE);
}

// ---------------------------------------------------------------------------
// Edge gather: wg[b,n,k] = mean_h attn[b,h,j,n] recomputed from q,k,bias and
// saved per-row softmax stats; ef[b,n,k,:] += wg * x[b,j,:]
// ---------------------------------------------------------------------------
__global__ void edge_gather_kernel(const float* __restrict__ q, const float* __restrict__ k,
                                   const float* __restrict__ bias, const float* __restrict__ Mrow,
                                   const float* __restrict__ Lrow, const int* __restrict__ eidx,
                                   const float* __restrict__ x, float* __restrict__ ef)
{
    const int t = blockIdx.x * 256 + threadIdx.x;
    if (t >= BB * NN * KKE) return;
    const int b = t / (NN * KKE);
    const int n = (t - b * NN * KKE) / KKE;
    const int j = eidx[t];

    const float bval = bias[((long)(b * NN + j)) * NN + n];
    float wg = 0.0f;
    for (int h = 0; h < HH; ++h) {
        const float* qp = q + ((long)(b * NN + j)) * DD + h * HDIM;
        const float* kp = k + ((long)(b * NN + n)) * DD + h * HDIM;
        float dot = 0.0f;
#pragma unroll
        for (int d = 0; d < HDIM; ++d) dot += qp[d] * kp[d];
        const float sc = dot + bval;
        wg += expf(sc - Mrow[(b * HH + h) * NN + j]) / Lrow[(b * HH + h) * NN + j];
    }
    wg *= (1.0f / HH);

    const float* xp = x + ((long)(b * NN + j)) * DD;
    float* efp = ef + (long)t * DD;
    for (int d = 0; d < DD; ++d) efp[d] += wg * xp[d];
}

// ---------------------------------------------------------------------------
// LayerNorm over last dim (Dn = 128 or 64), one wave per row.
// out = LN(in [+resid]) * gam + bet ; act=1 -> GELU after LN.
// ---------------------------------------------------------------------------
__global__ __launch_bounds__(32)
void ln_kernel(const float* __restrict__ in, const float* __restrict__ resid,
               const float* __restrict__ gam, const float* __restrict__ bet,
               float* __restrict__ out, int Dn, int act)
{
    const long row = blockIdx.x;
    const int lane = threadIdx.x;
    const int cnt = Dn >> 5;
    float t[4];
    const float* ip = in + row * Dn;
    float s = 0.0f;
    for (int i = 0; i < cnt; ++i) {
        const int c = lane + 32 * i;
        float x = ip[c];
        if (resid) x += resid[row * Dn + c];
        t[i] = x;
        s += x;
    }
#pragma unroll
    for (int off = 16; off >= 1; off >>= 1) s += __shfl_xor(s, off);
    const float mean = s / (float)Dn;
    float s2 = 0.0f;
    for (int i = 0; i < cnt; ++i) { const float d = t[i] - mean; s2 += d * d; }
#pragma unroll
    for (int off = 16; off >= 1; off >>= 1) s2 += __shfl_xor(s2, off);
    const float rstd = rsqrtf(s2 / (float)Dn + LN_EPS);
    for (int i = 0; i < cnt; ++i) {
        const int c = lane + 32 * i;
        float y = (t[i] - mean) * rstd * gam[c] + bet[c];
        if (act) y = gelu_f(y);
        out[row * Dn + c] = y;
    }
}

// ---------------------------------------------------------------------------
extern "C" void kernel_launch(void* const* d_in, const int* in_sizes, int n_in,
                              void* d_out, int out_size, void* d_ws, size_t ws_size,
                              hipStream_t stream)
{
    (void)in_sizes; (void)out_size; (void)ws_size;
    if (n_in < 27) return;

    const float* x_in  = (const float*)d_in[0];
    const float* e_in  = (const float*)d_in[1];
    const int*   eidx  = (const int*)  d_in[2];
    const float* Wq  = (const float*)d_in[3];  const float* bq  = (const float*)d_in[4];
    const float* Wk  = (const float*)d_in[5];  const float* bk  = (const float*)d_in[6];
    const float* Wv  = (const float*)d_in[7];  const float* bv  = (const float*)d_in[8];
    const float* Wo  = (const float*)d_in[9];  const float* bo  = (const float*)d_in[10];
    const float* W1  = (const float*)d_in[11]; const float* b1  = (const float*)d_in[12];
    const float* W2  = (const float*)d_in[13]; const float* b2  = (const float*)d_in[14];
    const float* Wef = (const float*)d_in[15]; const float* bef = (const float*)d_in[16];
    const float* Web = (const float*)d_in[17]; const float* beb = (const float*)d_in[18];
    const float* ln1s = (const float*)d_in[19]; const float* ln1b = (const float*)d_in[20];
    const float* ln2s = (const float*)d_in[21]; const float* ln2b = (const float*)d_in[22];
    const float* ln3s = (const float*)d_in[23]; const float* ln3b = (const float*)d_in[24];
    const float* ln4s = (const float*)d_in[25]; const float* ln4b = (const float*)d_in[26];

    const size_t XN  = (size_t)BB * NN * DD;         // 524288
    const size_t EN  = (size_t)BB * NN * KKE * DE;   // 4194304
    const size_t BN  = (size_t)BB * NN * NN;         // 2097152
    const size_t EFN = (size_t)BB * NN * KKE * DD;   // 8388608
    const size_t FN  = (size_t)BB * NN * FF;         // 1048576
    const size_t STAT = (size_t)BB * HH * NN;        // 32768
    const int BNK = BB * NN * KKE;                   // 65536

    float* ws = (float*)d_ws;
    size_t off = 0;
    float* xA   = ws + off; off += XN;
    float* xB   = ws + off; off += XN;
    float* eA   = ws + off; off += EN;
    float* eB   = ws + off; off += EN;
    float* bias = ws + off; off += BN;
    float* qb   = ws + off; off += XN;
    float* kb   = ws + off; off += XN;
    float* vb   = ws + off; off += XN;
    float* ob   = ws + off; off += XN;
    float* osum = ws + off; off += XN;
    float* hbuf = ws + off; off += XN;
    float* ffn1 = ws + off; off += FN;
    float* ffn2 = ws + off; off += XN;
    float* efb  = ws + off; off += EFN;
    float* etmp = ws + off; off += EN;
    float* Mr   = ws + off; off += STAT;
    float* Lr   = ws + off; off += STAT;

    hipMemcpyAsync(xA, x_in, XN * sizeof(float), hipMemcpyDeviceToDevice, stream);
    hipMemcpyAsync(eA, e_in, EN * sizeof(float), hipMemcpyDeviceToDevice, stream);

    float *xc = xA, *xn = xB, *ec = eA, *en = eB;
    const dim3 blk(32);

    for (int l = 0; l < LLAY; ++l) {
        const float* Wq_l  = Wq  + (size_t)l * DD * DD;  const float* bq_l  = bq  + (size_t)l * DD;
        const float* Wk_l  = Wk  + (size_t)l * DD * DD;  const float* bk_l  = bk  + (size_t)l * DD;
        const float* Wv_l  = Wv  + (size_t)l * DD * DD;  const float* bv_l  = bv  + (size_t)l * DD;
        const float* Wo_l  = Wo  + (size_t)l * DD * DD;  const float* bo_l  = bo  + (size_t)l * DD;
        const float* W1_l  = W1  + (size_t)l * DD * FF;  const float* b1_l  = b1  + (size_t)l * FF;
        const float* W2_l  = W2  + (size_t)l * FF * DD;  const float* b2_l  = b2  + (size_t)l * DD;
        const float* Wef_l = Wef + (size_t)l * DE * DD;  const float* bef_l = bef + (size_t)l * DD;
        const float* Web_l = Web + (size_t)l * DD * DE;  const float* beb_l = beb + (size_t)l * DE;
        const float* ln1s_l = ln1s + (size_t)l * DD; const float* ln1b_l = ln1b + (size_t)l * DD;
        const float* ln2s_l = ln2s + (size_t)l * DD; const float* ln2b_l = ln2b + (size_t)l * DD;
        const float* ln3s_l = ln3s + (size_t)l * DD; const float* ln3b_l = ln3b + (size_t)l * DD;
        const float* ln4s_l = ln4s + (size_t)l * DE; const float* ln4b_l = ln4b + (size_t)l * DE;

        // attn_bias
        hipMemsetAsync(bias, 0, BN * sizeof(float), stream);
        bias_scatter_kernel<<<(BNK + 255) / 256, 256, 0, stream>>>(ec, eidx, bias);

        // QKV projections (q pre-scaled by HD^-0.5 = 0.25)
        wmma_gemm_kernel<<<dim3(256, 8), blk, 0, stream>>>(xc, Wq_l, bq_l, nullptr, qb, DD, DD, 0.25f, 0);
        wmma_gemm_kernel<<<dim3(256, 8), blk, 0, stream>>>(xc, Wk_l, bk_l, nullptr, kb, DD, DD, 1.0f, 0);
        wmma_gemm_kernel<<<dim3(256, 8), blk, 0, stream>>>(xc, Wv_l, bv_l, nullptr, vb, DD, DD, 1.0f, 0);

        // flash attention + per-row stats
        attn_flash_kernel<<<dim3(NN / 16, HH, BB), blk, 0, stream>>>(qb, kb, vb, bias, ob, Mr, Lr);

        // O projection + residual, then LN1 -> h
        wmma_gemm_kernel<<<dim3(256, 8), blk, 0, stream>>>(ob, Wo_l, bo_l, xc, osum, DD, DD, 1.0f, 0);
        ln_kernel<<<BB * NN, blk, 0, stream>>>(osum, nullptr, ln1s_l, ln1b_l, hbuf, DD, 0);

        // FFN: gelu(h@W1+b1)@W2+b2 + h, then LN3 -> x_new
        wmma_gemm_kernel<<<dim3(256, 16), blk, 0, stream>>>(hbuf, W1_l, b1_l, nullptr, ffn1, DD, FF, 1.0f, 1);
        wmma_gemm_kernel<<<dim3(256, 8),  blk, 0, stream>>>(ffn1, W2_l, b2_l, hbuf, ffn2, FF, DD, 1.0f, 0);
        ln_kernel<<<BB * NN, blk, 0, stream>>>(ffn2, nullptr, ln3s_l, ln3b_l, xn, DD, 0);

        // Edge path: ef = e@Wef+bef ; ef += wg*xg ; gelu(LN2(ef)) ; e_new = LN4(e + ef@Web+beb)
        wmma_gemm_kernel<<<dim3(4096, 8), blk, 0, stream>>>(ec, Wef_l, bef_l, nullptr, efb, DE, DD, 1.0f, 0);
        edge_gather_kernel<<<(BNK + 255) / 256, 256, 0, stream>>>(qb, kb, bias, Mr, Lr, eidx, xc, efb);
        ln_kernel<<<BNK, blk, 0, stream>>>(efb, nullptr, ln2s_l, ln2b_l, efb, DD, 1);
        wmma_gemm_kernel<<<dim3(4096, 4), blk, 0, stream>>>(efb, Web_l, beb_l, ec, etmp, DD, DE, 1.0f, 0);
        ln_kernel<<<BNK, blk, 0, stream>>>(etmp, nullptr, ln4s_l, ln4b_l, en, DE, 0);

        // ping-pong
        float* tx = xc; xc = xn; xn = tx;
        float* te = ec; ec = en; en = te;
    }

    hipMemcpyAsync(d_out, xc, XN * sizeof(float), hipMemcpyDeviceToDevice, stream);
    hipMemcpyAsync((float*)d_out + XN, ec, EN * sizeof(float), hipMemcpyDeviceToDevice, stream);
}